// FAGCN_82231443849289
// MI455X (gfx1250) — compile-verified
//
#include <hip/hip_runtime.h>
#include <math.h>

#define N_NODES 50000
#define N_EDGES 800000
#define E_TOT   (N_EDGES + N_NODES)   // edges + self loops = 850000
#define F_IN    256
#define F_HID   64
#define F_OUT   16
#define EPS_F   0.1f

typedef __attribute__((ext_vector_type(2))) float v2f;
typedef __attribute__((ext_vector_type(8))) float v8f;

// ---------------- utility ----------------
__global__ void fill_kernel(float* __restrict__ p, float val, int n) {
    int i = blockIdx.x * blockDim.x + threadIdx.x;
    int stride = gridDim.x * blockDim.x;
    for (; i < n; i += stride) p[i] = val;
}

// deg starts at 1.0 (self loop); add 1 per incoming edge
__global__ void deg_kernel(const int* __restrict__ col, float* __restrict__ deg) {
    int e = blockIdx.x * blockDim.x + threadIdx.x;
    if (e < N_EDGES) atomicAdd(&deg[col[e]], 1.0f);
}

__global__ void dinv_kernel(float* __restrict__ deg) {
    int i = blockIdx.x * blockDim.x + threadIdx.x;
    if (i < N_NODES) deg[i] = rsqrtf(deg[i]);   // deg >= 1 always
}

__global__ void norm_kernel(const int* __restrict__ row, const int* __restrict__ col,
                            const float* __restrict__ dinv, float* __restrict__ nrm) {
    int e = blockIdx.x * blockDim.x + threadIdx.x;
    if (e >= E_TOT) return;
    float n;
    if (e < N_EDGES) {
        n = dinv[row[e]] * dinv[col[e]];
    } else {
        float d = dinv[e - N_EDGES];
        n = d * d;
    }
    nrm[e] = n;
}

// ---------------- GEMM1: h = relu(x @ W1 + b1), [50000,256]x[256,64] ----------------
// W1 staged once per block into LDS in WMMA-B pair layout:
//   lds index ((kstep*2 + half)*4 + nt)*16 + n  holds pair (W1[k0,n'], W1[k0+1,n'])
//   with k0 = kstep*4 + half*2, n' = nt*16 + n.
// Inner loop: one ds_load_b64 per N-tile per k-step; A streamed from global.
__global__ void gemm1_kernel(const float* __restrict__ x, const float* __restrict__ W1,
                             const float* __restrict__ b1, float* __restrict__ h) {
    __shared__ v2f lw[(F_IN / 4) * 2 * 4 * 16];   // 8192 pairs = 64KB

    // cooperative stage + swizzle of W1
    for (int p = threadIdx.x; p < 8192; p += blockDim.x) {
        int kstep = p >> 7;            // 128 entries per kstep
        int rem   = p & 127;
        int half  = rem >> 6;
        int nt    = (rem >> 4) & 3;
        int n     = rem & 15;
        int k0    = kstep * 4 + half * 2;
        v2f w;
        w.x = W1[(size_t)k0 * F_HID + nt * 16 + n];
        w.y = W1[(size_t)(k0 + 1) * F_HID + nt * 16 + n];
        lw[p] = w;
    }
    __syncthreads();

    int wave = (blockIdx.x * blockDim.x + threadIdx.x) >> 5;   // uniform within wave
    int lane = threadIdx.x & 31;
    if (wave >= N_NODES / 16) return;

    int m0    = wave * 16;
    int arow  = m0 + (lane & 15);         // A-frag rows
    int koff  = (lane >> 4) << 1;         // lanes 0-15 -> K 0,1 ; lanes 16-31 -> K 2,3
    int half  = lane >> 4;
    int nlane = lane & 15;

    v8f acc0 = {}, acc1 = {}, acc2 = {}, acc3 = {};
    const float* xr = x + (size_t)arow * F_IN;

    for (int k = 0; k < F_IN; k += 4) {
        v2f a = *(const v2f*)(xr + k + koff);
        int base = (((k >> 2) * 2 + half) * 4) * 16 + nlane;
        v2f b0  = lw[base];
        v2f b1v = lw[base + 16];
        v2f b2v = lw[base + 32];
        v2f b3v = lw[base + 48];
        acc0 = __builtin_amdgcn_wmma_f32_16x16x4_f32(false, a, false, b0,  (short)0, acc0, false, false);
        acc1 = __builtin_amdgcn_wmma_f32_16x16x4_f32(false, a, false, b1v, (short)0, acc1, false, false);
        acc2 = __builtin_amdgcn_wmma_f32_16x16x4_f32(false, a, false, b2v, (short)0, acc2, false, false);
        acc3 = __builtin_amdgcn_wmma_f32_16x16x4_f32(false, a, false, b3v, (short)0, acc3, false, false);
    }

    // C/D layout: VGPR v holds row m0 + v (lanes 0-15) or m0 + 8 + v (lanes 16-31)
    int mbase = m0 + (half << 3);
    v8f accs[4] = {acc0, acc1, acc2, acc3};
#pragma unroll
    for (int nt = 0; nt < 4; ++nt) {
        float bias = b1[nt * 16 + nlane];
#pragma unroll
        for (int v = 0; v < 8; ++v) {
            float val = accs[nt][v] + bias;
            h[(size_t)(mbase + v) * F_HID + nt * 16 + nlane] = fmaxf(val, 0.0f);
        }
    }
}

// ---------------- gate factor: sigmoid(h . att_l)*(1+eps) - eps ----------------
__global__ void factor_kernel(const float* __restrict__ h, const float* __restrict__ att_l,
                              float* __restrict__ factor) {
    int node = (blockIdx.x * blockDim.x + threadIdx.x) >> 5;
    int lane = threadIdx.x & 31;
    if (node >= N_NODES) return;
    v2f hv = *(const v2f*)(h + (size_t)node * F_HID + lane * 2);
    v2f av = *(const v2f*)(att_l + lane * 2);
    float p = hv.x * av.x + hv.y * av.y;
#pragma unroll
    for (int off = 16; off > 0; off >>= 1) p += __shfl_xor(p, off, 32);
    if (lane == 0) {
        float gate = 1.0f / (1.0f + __expf(-p));
        factor[node] = gate * (1.0f + EPS_F) - EPS_F;
    }
}

// ---------------- edge scatter: xl[col] += norm * h[row]  (one wave / edge) ----------------
__global__ void scatter_kernel(const float* __restrict__ hsrc, const int* __restrict__ row,
                               const int* __restrict__ col, const float* __restrict__ nrm,
                               float* __restrict__ xl) {
    int tid  = blockIdx.x * blockDim.x + threadIdx.x;
    int lane = threadIdx.x & 31;
    int e    = tid >> 5;
    if (e >= E_TOT) return;
    int r, c;
    if (e < N_EDGES) { r = row[e]; c = col[e]; }
    else             { r = e - N_EDGES; c = r; }     // self loop
    float w = nrm[e];
    v2f v = *(const v2f*)(hsrc + (size_t)r * F_HID + lane * 2);
    float* dst = xl + (size_t)c * F_HID + lane * 2;
    atomicAdd(dst,     w * v.x);
    atomicAdd(dst + 1, w * v.y);
}

// ---------------- in-place rescale: h[i,f] = factor[i] * xl[i,f] ----------------
__global__ void scale_kernel(float* __restrict__ xl, const float* __restrict__ factor) {
    int i = blockIdx.x * blockDim.x + threadIdx.x;
    if (i < N_NODES * F_HID) xl[i] *= factor[i >> 6];
}

// ---------------- GEMM2: out = h @ W2 + b2, [50000,64]x[64,16] ----------------
__global__ void gemm2_kernel(const float* __restrict__ h, const float* __restrict__ W2,
                             const float* __restrict__ b2, float* __restrict__ out) {
    __shared__ v2f lw2[(F_HID / 4) * 2 * 16];   // 512 pairs = 4KB

    for (int p = threadIdx.x; p < 512; p += blockDim.x) {
        int kstep = p >> 5;            // 32 entries per kstep
        int rem   = p & 31;
        int half  = rem >> 4;
        int n     = rem & 15;
        int k0    = kstep * 4 + half * 2;
        v2f w;
        w.x = W2[k0 * F_OUT + n];
        w.y = W2[(k0 + 1) * F_OUT + n];
        lw2[p] = w;
    }
    __syncthreads();

    int wave = (blockIdx.x * blockDim.x + threadIdx.x) >> 5;
    int lane = threadIdx.x & 31;
    if (wave >= N_NODES / 16) return;
    int m0    = wave * 16;
    int arow  = m0 + (lane & 15);
    int koff  = (lane >> 4) << 1;
    int half  = lane >> 4;
    int nlane = lane & 15;
    v8f acc = {};
    const float* hr = h + (size_t)arow * F_HID;
#pragma unroll
    for (int k = 0; k < F_HID; k += 4) {
        v2f a = *(const v2f*)(hr + k + koff);
        v2f b = lw2[((k >> 2) * 2 + half) * 16 + nlane];
        acc = __builtin_amdgcn_wmma_f32_16x16x4_f32(false, a, false, b, (short)0, acc, false, false);
    }
    int mbase = m0 + (half << 3);
    float bias = b2[nlane];
#pragma unroll
    for (int v = 0; v < 8; ++v)
        out[(size_t)(mbase + v) * F_OUT + nlane] = acc[v] + bias;
}

// ---------------- driver ----------------
extern "C" void kernel_launch(void* const* d_in, const int* in_sizes, int n_in,
                              void* d_out, int out_size, void* d_ws, size_t ws_size,
                              hipStream_t stream) {
    const float* x   = (const float*)d_in[0];
    const int*   ei  = (const int*)d_in[1];     // [2, E] (integer -> int32 per harness)
    const float* W1  = (const float*)d_in[2];
    const float* b1  = (const float*)d_in[3];
    const float* att = (const float*)d_in[4];   // [2,1,64]
    const float* W2  = (const float*)d_in[5];
    const float* b2  = (const float*)d_in[6];
    float*       out = (float*)d_out;

    const int* row = ei;            // edge_index[0]
    const int* col = ei + N_EDGES;  // edge_index[1]

    char* ws = (char*)d_ws;
    size_t o = 0;
    auto take = [&](size_t nfloats) -> float* {
        float* p = (float*)(ws + o);
        o += ((nfloats * sizeof(float) + 255) & ~(size_t)255);
        return p;
    };
    float* dinv   = take(N_NODES);
    float* nrm    = take(E_TOT);
    float* factor = take(N_NODES);
    float* h0     = take((size_t)N_NODES * F_HID);
    float* h1     = take((size_t)N_NODES * F_HID);

    const int B = 256;
    const int nf = N_NODES * F_HID;
    const int mtiles = N_NODES / 16;                       // 3125 exact
    const int gemm_blocks = (mtiles + (B / 32) - 1) / (B / 32);
    const long long scat_threads = (long long)E_TOT * 32;
    const int scat_blocks = (int)((scat_threads + B - 1) / B);
    const int fac_blocks = (int)(((long long)N_NODES * 32 + B - 1) / B);

    // degree -> dinv -> per-edge norm
    fill_kernel<<<512, B, 0, stream>>>(dinv, 1.0f, N_NODES);           // self-loop contribution
    deg_kernel<<<(N_EDGES + B - 1) / B, B, 0, stream>>>(col, dinv);
    dinv_kernel<<<(N_NODES + B - 1) / B, B, 0, stream>>>(dinv);
    norm_kernel<<<(E_TOT + B - 1) / B, B, 0, stream>>>(row, col, dinv, nrm);

    // h0 = relu(x @ W1 + b1)
    gemm1_kernel<<<gemm_blocks, B, 0, stream>>>(x, W1, b1, h0);

    // layer 0: h1 = factor(h0) * scatter(h0)
    fill_kernel<<<2048, B, 0, stream>>>(h1, 0.0f, nf);
    factor_kernel<<<fac_blocks, B, 0, stream>>>(h0, att, factor);
    scatter_kernel<<<scat_blocks, B, 0, stream>>>(h0, row, col, nrm, h1);
    scale_kernel<<<(nf + B - 1) / B, B, 0, stream>>>(h1, factor);

    // layer 1: h0 = factor(h1) * scatter(h1)
    fill_kernel<<<2048, B, 0, stream>>>(h0, 0.0f, nf);
    factor_kernel<<<fac_blocks, B, 0, stream>>>(h1, att + F_HID, factor);
    scatter_kernel<<<scat_blocks, B, 0, stream>>>(h1, row, col, nrm, h0);
    scale_kernel<<<(nf + B - 1) / B, B, 0, stream>>>(h0, factor);

    // out = h0 @ W2 + b2
    gemm2_kernel<<<gemm_blocks, B, 0, stream>>>(h0, W2, b2, out);
}